// LNN_49417893707899
// MI455X (gfx1250) — compile-verified
//
#include <hip/hip_runtime.h>

// LNN acceleration on gfx1250: analytic gradient + Hessian v-rows of a
// 3-layer softplus MLP; all 512-dim contractions via V_WMMA_F32_16X16X4_F32.
// 16 samples/WG; Hessian passes process 4 sample-pairs per sweep of W2
// (B tiles reused 4x) to minimize L2 traffic. Activation buffers are
// minimized (sigmoid(z2) and u only) so the 4-pair sT fits in 320KB LDS.

typedef float v2f __attribute__((ext_vector_type(2)));
typedef float v8f __attribute__((ext_vector_type(8)));

#define HDIM 512
#define HP   516      // padded row stride: 4-bank skew, float2-aligned
#define DDIM 16
#define SPB  16       // samples per block
#define NPASS 2       // 8 pairs, 4 per pass
#define PPP  4        // pairs per pass
#define NWAVE 8
#define NTHREADS 256
#define NTILE 4       // 64-wide column slice per wave

__device__ __forceinline__ v8f wmma_f32_k4(v2f a, v2f b, v8f c) {
  return __builtin_amdgcn_wmma_f32_16x16x4_f32(false, a, false, b, (short)0, c,
                                               false, false);
}

__global__ __launch_bounds__(NTHREADS)
void lnn_accel_kernel(const float* __restrict__ x,
                      const float* __restrict__ v,
                      const float* __restrict__ W1,
                      const float* __restrict__ b1,
                      const float* __restrict__ W2,
                      const float* __restrict__ b2,
                      const float* __restrict__ W3,
                      float* __restrict__ out)
{
  __shared__ float sW1[DDIM][HP];     // row-major W1, padded       33.0 KB
  __shared__ float sW3[HDIM];
  __shared__ float sB1[HDIM];
  __shared__ float sB2[HDIM];
  __shared__ float sS1[SPB][HP];      // sigmoid(z1)                33.0 KB
  __shared__ float sH1[SPB][HP];      // softplus(z1) -> later u    33.0 KB
  __shared__ float sZ2[SPB][HP];      // sigmoid(z2)                33.0 KB
  __shared__ float sT [PPP][16][HP];  // 4 pairs: Tv*s2, then F    132.1 KB
  __shared__ float sDyL[SPB][DDIM];
  __shared__ float sHp[NWAVE][PPP][16][16];
  __shared__ float sHv[SPB/2][16][16];
  // total ~305 KB

  const int tid  = threadIdx.x;
  const int lane = tid & 31;
  const int wave = tid >> 5;
  const int colBase = wave * 64;        // this wave's 64-wide output slice
  const int s0 = blockIdx.x * SPB;

  const int arow = lane & 15;           // WMMA row (A) or column (B/D)
  const int koff = (lane >> 4) * 2;     // K offset within 4-wide K tile
  const int rsel = (lane >> 4) << 3;    // D-layout row offset (0 or 8)

  // ---------------- stage 0: stage weights into LDS ----------------
  {
    const float4* src = (const float4*)W1;
    for (int i = tid; i < (DDIM * HDIM) / 4; i += NTHREADS) {
      const int row = i / (HDIM / 4), c4 = i % (HDIM / 4);
      *(float4*)(&sW1[row][c4 * 4]) = src[i];
    }
    for (int i = tid; i < HDIM; i += NTHREADS) {
      sW3[i] = W3[i];
      sB1[i] = b1[i];
      sB2[i] = b2[i];
    }
  }
  __syncthreads();

  // ---------------- stage 1: z1, softplus, sigmoid ----------------
  {
    const int s  = tid >> 4;            // 16 samples, 16 threads each
    const int h0 = (tid & 15) * 32;
    float y[16];
    #pragma unroll
    for (int d = 0; d < 8; ++d) y[d]     = x[(size_t)(s0 + s) * 8 + d];
    #pragma unroll
    for (int d = 0; d < 8; ++d) y[8 + d] = v[(size_t)(s0 + s) * 8 + d];
    for (int h = h0; h < h0 + 32; ++h) {
      float z = sB1[h];
      #pragma unroll
      for (int d = 0; d < 16; ++d) z = fmaf(y[d], sW1[d][h], z);
      float sp = fmaxf(z, 0.f) + log1pf(expf(-fabsf(z)));
      float sg = 1.f / (1.f + expf(-z));
      sH1[s][h] = sp;
      sS1[s][h] = sg;
    }
  }
  __syncthreads();

  // ---------------- stage 2: z2 = h1 @ W2 -> sigmoid(z2) ----------------
  {
    v8f acc[NTILE];
    #pragma unroll
    for (int t = 0; t < NTILE; ++t)
      #pragma unroll
      for (int r = 0; r < 8; ++r) acc[t][r] = 0.f;

    for (int k = 0; k < HDIM; k += 4) {
      const float2 af = *(const float2*)(&sH1[arow][k + koff]);
      v2f a; a.x = af.x; a.y = af.y;
      #pragma unroll
      for (int t = 0; t < NTILE; ++t) {
        const int j = colBase + t * 16 + arow;
        v2f b;
        b.x = W2[(size_t)(k + koff    ) * HDIM + j];
        b.y = W2[(size_t)(k + koff + 1) * HDIM + j];
        acc[t] = wmma_f32_k4(a, b, acc[t]);
      }
    }
    #pragma unroll
    for (int t = 0; t < NTILE; ++t) {
      const int j = colBase + t * 16 + arow;
      #pragma unroll
      for (int r = 0; r < 8; ++r) {
        const float zz = acc[t][r] + sB2[j];
        sZ2[r + rsel][j] = 1.f / (1.f + expf(-zz));
      }
    }
  }
  __syncthreads();

  // ------------- stage 3: u = (sigmoid(z2)*W3) @ W2^T -> u into sH1 -------
  {
    v8f acc[NTILE];
    #pragma unroll
    for (int t = 0; t < NTILE; ++t)
      #pragma unroll
      for (int r = 0; r < 8; ++r) acc[t][r] = 0.f;

    for (int k = 0; k < HDIM; k += 4) {
      const float2 zf = *(const float2*)(&sZ2[arow][k + koff]);
      const float2 wf = *(const float2*)(&sW3[k + koff]);   // broadcast
      v2f a; a.x = zf.x * wf.x; a.y = zf.y * wf.y;          // g2 on the fly
      #pragma unroll
      for (int t = 0; t < NTILE; ++t) {
        const int h = colBase + t * 16 + arow;
        const float2 bf = *(const float2*)(W2 + (size_t)h * HDIM + k + koff);
        v2f b; b.x = bf.x; b.y = bf.y;
        acc[t] = wmma_f32_k4(a, b, acc[t]);
      }
    }
    #pragma unroll
    for (int t = 0; t < NTILE; ++t) {
      const int h = colBase + t * 16 + arow;
      #pragma unroll
      for (int r = 0; r < 8; ++r) sH1[r + rsel][h] = acc[t][r];  // u
    }
  }
  __syncthreads();

  // ---------------- stage 4: dL/dy = (s1*u) @ W1^T via WMMA ----------------
  {
    v8f acc;
    #pragma unroll
    for (int r = 0; r < 8; ++r) acc[r] = 0.f;
    for (int k = colBase; k < colBase + 64; k += 4) {
      const float2 uf = *(const float2*)(&sH1[arow][k + koff]);
      const float2 sf = *(const float2*)(&sS1[arow][k + koff]);
      const float2 bf = *(const float2*)(&sW1[arow][k + koff]);  // W1^T tile
      v2f a; a.x = uf.x * sf.x; a.y = uf.y * sf.y;               // g1
      v2f b; b.x = bf.x; b.y = bf.y;
      acc = wmma_f32_k4(a, b, acc);
    }
    #pragma unroll
    for (int r = 0; r < 8; ++r) sHp[wave][0][r + rsel][arow] = acc[r];
  }
  __syncthreads();
  {
    const int rr = tid >> 4, cc = tid & 15;   // 256 entries, one per thread
    float s = 0.f;
    #pragma unroll
    for (int w = 0; w < NWAVE; ++w) s += sHp[w][0][rr][cc];
    sDyL[rr][cc] = s;
  }
  __syncthreads();

  // ---------------- stage 5: Hessian v-rows, 4 pairs per W2 sweep ---------
  for (int pp = 0; pp < NPASS; ++pp) {
    const int w1r = 8 + (arow & 7);

    // Tv = (W1v*s1) @ W2 for 4 pairs, scaled by s2 -> sT
    {
      v8f acc[PPP][NTILE];
      #pragma unroll
      for (int q = 0; q < PPP; ++q)
        #pragma unroll
        for (int t = 0; t < NTILE; ++t)
          #pragma unroll
          for (int r = 0; r < 8; ++r) acc[q][t][r] = 0.f;

      for (int k = 0; k < HDIM; k += 4) {
        const float2 wf = *(const float2*)(&sW1[w1r][k + koff]);
        v2f a[PPP];
        #pragma unroll
        for (int q = 0; q < PPP; ++q) {
          const int sq = 8 * pp + 2 * q + (arow >> 3);
          const float2 f = *(const float2*)(&sS1[sq][k + koff]);
          a[q].x = wf.x * f.x; a[q].y = wf.y * f.y;
        }
        #pragma unroll
        for (int t = 0; t < NTILE; ++t) {
          const int j = colBase + t * 16 + arow;
          v2f b;
          b.x = W2[(size_t)(k + koff    ) * HDIM + j];
          b.y = W2[(size_t)(k + koff + 1) * HDIM + j];
          #pragma unroll
          for (int q = 0; q < PPP; ++q)       // B tile reused 4x
            acc[q][t] = wmma_f32_k4(a[q], b, acc[q][t]);
        }
      }
      #pragma unroll
      for (int q = 0; q < PPP; ++q)
        #pragma unroll
        for (int t = 0; t < NTILE; ++t) {
          const int j = colBase + t * 16 + arow;
          const float w3j = sW3[j];
          #pragma unroll
          for (int r = 0; r < 8; ++r) {
            const int rr = r + rsel;
            const int sr = 8 * pp + 2 * q + (rr >> 3);
            const float sg2 = sZ2[sr][j];
            sT[q][rr][j] = acc[q][t][r] * sg2 * (1.f - sg2) * w3j;
          }
        }
    }
    __syncthreads();

    // Vv = Ts @ W2^T for 4 pairs (contract over j)
    v8f accV[PPP][NTILE];
    #pragma unroll
    for (int q = 0; q < PPP; ++q)
      #pragma unroll
      for (int t = 0; t < NTILE; ++t)
        #pragma unroll
        for (int r = 0; r < 8; ++r) accV[q][t][r] = 0.f;

    for (int k = 0; k < HDIM; k += 4) {
      v2f a[PPP];
      #pragma unroll
      for (int q = 0; q < PPP; ++q) {
        const float2 af = *(const float2*)(&sT[q][arow][k + koff]);
        a[q].x = af.x; a[q].y = af.y;
      }
      #pragma unroll
      for (int t = 0; t < NTILE; ++t) {
        const int h = colBase + t * 16 + arow;
        const float2 bf = *(const float2*)(W2 + (size_t)h * HDIM + k + koff);
        v2f b; b.x = bf.x; b.y = bf.y;
        #pragma unroll
        for (int q = 0; q < PPP; ++q)
          accV[q][t] = wmma_f32_k4(a[q], b, accV[q][t]);
      }
    }
    __syncthreads();                 // all waves done reading sT

    // F = Vv*s1 + W1v*(s1(1-s1)*u) -> overwrite sT
    #pragma unroll
    for (int q = 0; q < PPP; ++q)
      #pragma unroll
      for (int t = 0; t < NTILE; ++t) {
        const int h = colBase + t * 16 + arow;
        #pragma unroll
        for (int r = 0; r < 8; ++r) {
          const int rr = r + rsel;
          const int sr = 8 * pp + 2 * q + (rr >> 3);
          const float s1v = sS1[sr][h];
          const float uv  = sH1[sr][h];
          const float w1h = sW1[8 + (rr & 7)][h];
          sT[q][rr][h] = accV[q][t][r] * s1v + w1h * (s1v * (1.f - s1v) * uv);
        }
      }
    __syncthreads();

    // Hv partials = F_slice @ W1^T, 4 pairs
    {
      v8f accH[PPP];
      #pragma unroll
      for (int q = 0; q < PPP; ++q)
        #pragma unroll
        for (int r = 0; r < 8; ++r) accH[q][r] = 0.f;
      for (int k = colBase; k < colBase + 64; k += 4) {
        const float2 bf = *(const float2*)(&sW1[arow][k + koff]);
        v2f b; b.x = bf.x; b.y = bf.y;
        #pragma unroll
        for (int q = 0; q < PPP; ++q) {
          const float2 af = *(const float2*)(&sT[q][arow][k + koff]);
          v2f a; a.x = af.x; a.y = af.y;
          accH[q] = wmma_f32_k4(a, b, accH[q]);
        }
      }
      #pragma unroll
      for (int q = 0; q < PPP; ++q)
        #pragma unroll
        for (int r = 0; r < 8; ++r)
          sHp[wave][q][r + rsel][arow] = accH[q][r];
    }
    __syncthreads();

    // reduce 8 wave partials -> sHv[4*pp + q]
    for (int e = tid; e < PPP * 256; e += NTHREADS) {
      const int qq = e >> 8, rr2 = (e >> 4) & 15, cc = e & 15;
      float s = 0.f;
      #pragma unroll
      for (int w = 0; w < NWAVE; ++w) s += sHp[w][qq][rr2][cc];
      sHv[PPP * pp + qq][rr2][cc] = s;
    }
    __syncthreads();
  }

  // ---------------- stage 6: rhs + 8x8 solve (Gauss-Jordan) ----------------
  if (tid < SPB) {
    const int s = tid, p = s >> 1, r0 = (s & 1) * 8;
    float vv[8];
    #pragma unroll
    for (int j = 0; j < 8; ++j) vv[j] = v[(size_t)(s0 + s) * 8 + j];
    float M[8][9];
    for (int i = 0; i < 8; ++i) {
      for (int j = 0; j < 8; ++j) M[i][j] = sHv[p][r0 + i][8 + j];  // dvdv
      float r = sDyL[s][i];                                         // dxL
      for (int j = 0; j < 8; ++j) r -= sHv[p][r0 + i][j] * vv[j];   // -dxdv@v
      M[i][8] = r;
    }
    for (int c = 0; c < 8; ++c) {
      int piv = c; float best = fabsf(M[c][c]);
      for (int r2 = c + 1; r2 < 8; ++r2) {
        float a = fabsf(M[r2][c]);
        if (a > best) { best = a; piv = r2; }
      }
      if (piv != c)
        for (int j = c; j < 9; ++j) { float t = M[c][j]; M[c][j] = M[piv][j]; M[piv][j] = t; }
      float inv = 1.f / M[c][c];
      for (int j = c; j < 9; ++j) M[c][j] *= inv;
      for (int r2 = 0; r2 < 8; ++r2) if (r2 != c) {
        float f = M[r2][c];
        for (int j = c; j < 9; ++j) M[r2][j] -= f * M[c][j];
      }
    }
    for (int i = 0; i < 8; ++i) out[(size_t)(s0 + s) * 8 + i] = M[i][8];
  }
}

extern "C" void kernel_launch(void* const* d_in, const int* in_sizes, int n_in,
                              void* d_out, int out_size, void* d_ws, size_t ws_size,
                              hipStream_t stream) {
  const float* x  = (const float*)d_in[0];
  const float* v  = (const float*)d_in[1];
  const float* W1 = (const float*)d_in[2];
  const float* b1 = (const float*)d_in[3];
  const float* W2 = (const float*)d_in[4];
  const float* b2 = (const float*)d_in[5];
  const float* W3 = (const float*)d_in[6];
  // d_in[7] = b3: drops out of all derivatives
  float* out = (float*)d_out;

  const int nB = in_sizes[0] / 8;          // 8192
  const int grid = nB / SPB;               // 512 workgroups
  lnn_accel_kernel<<<grid, NTHREADS, 0, stream>>>(x, v, W1, b1, W2, b2, W3, out);
}